// GCN_Original_64501818851896
// MI455X (gfx1250) — compile-verified
//
#include <hip/hip_runtime.h>
#include <hip/hip_bf16.h>

typedef __attribute__((ext_vector_type(2))) float v2f;
typedef __attribute__((ext_vector_type(8))) float v8f;

#define N_NODES  100000
#define D_FEAT   64
#define N_CH     64
#define N_EDGES  1600000

// ---------------------------------------------------------------------------
// Kernel 1: h = features @ W + bias   (fp32 WMMA 16x16x4)
// One wave computes a 16-row x 64-col slab: 4 accumulator tiles, 16 k-steps.
// W staged in LDS K-pair-interleaved so each B fragment is ONE ds_load_b64:
//   ldsW2[kpair*64 + col] = { W[2*kpair][col], W[2*kpair+1][col] }
// ---------------------------------------------------------------------------
__global__ __launch_bounds__(256) void gcn_gemm_wmma(
    const float* __restrict__ F, const float* __restrict__ W,
    const float* __restrict__ bias, float* __restrict__ H) {
  __shared__ v2f   ldsW2[(D_FEAT / 2) * N_CH];   // 16 KB
  __shared__ float ldsB[N_CH];
  for (int i = threadIdx.x; i < (D_FEAT / 2) * N_CH; i += 256) {
    const int kpair = i >> 6;        // 0..31
    const int col   = i & 63;
    v2f w;
    w.x = W[(2 * kpair + 0) * N_CH + col];
    w.y = W[(2 * kpair + 1) * N_CH + col];
    ldsW2[i] = w;
  }
  if (threadIdx.x < N_CH) ldsB[threadIdx.x] = bias[threadIdx.x];
  __syncthreads();

  const int wave = threadIdx.x >> 5;
  const int lane = threadIdx.x & 31;
  const int half = lane >> 4;   // 0: lanes 0-15, 1: lanes 16-31
  const int l16  = lane & 15;
  const int row0 = blockIdx.x * 128 + wave * 16;
  if (row0 >= N_NODES) return;  // N_NODES % 16 == 0 -> whole-wave uniform exit

  v8f acc[4] = {{}, {}, {}, {}};
  const float* frow = F + (size_t)(row0 + l16) * D_FEAT;

  for (int kk = 0; kk < D_FEAT; kk += 4) {
    // A fragment: lane-half selects K pair {kk,kk+1} vs {kk+2,kk+3}
    const int k2 = kk + 2 * half;
    v2f a;
    a.x = frow[k2];
    a.y = frow[k2 + 1];
    // B fragments: kpair index = k2/2 = kk/2 + half  -> single b64 DS load
    const v2f* wrow = ldsW2 + (size_t)((kk >> 1) + half) * N_CH;
#pragma unroll
    for (int ct = 0; ct < 4; ++ct) {
      v2f b = wrow[ct * 16 + l16];
      acc[ct] = __builtin_amdgcn_wmma_f32_16x16x4_f32(
          /*neg_a=*/false, a, /*neg_b=*/false, b,
          /*c_mod=*/(short)0, acc[ct], /*reuse_a=*/false, /*reuse_b=*/false);
    }
  }

#pragma unroll
  for (int ct = 0; ct < 4; ++ct) {
    const float bb = ldsB[ct * 16 + l16];
#pragma unroll
    for (int v = 0; v < 8; ++v) {
      H[(size_t)(row0 + v + 8 * half) * N_CH + ct * 16 + l16] = acc[ct][v] + bb;
    }
  }
}

// ---------------------------------------------------------------------------
// Kernel 2: zero the accumulator buffer (float4 stores)
// ---------------------------------------------------------------------------
__global__ __launch_bounds__(256) void zero_f4(float4* __restrict__ p, int n4) {
  int t = blockIdx.x * 256 + threadIdx.x;
  if (t < n4) p[t] = make_float4(0.f, 0.f, 0.f, 0.f);
}

// ---------------------------------------------------------------------------
// Kernel 3: edge-parallel SpMM scatter with native f32 atomics.
// One wave per edge; each lane handles 2 channels (float2 gather).
// Edge metadata forced wave-uniform via readfirstlane -> scalar (s_load) path.
// h and agg both fit in the 192MB L2, so this runs at L2 bandwidth.
// ---------------------------------------------------------------------------
__global__ __launch_bounds__(256) void spmm_scatter(
    const float* __restrict__ H, const float* __restrict__ vals,
    const int* __restrict__ rows, const int* __restrict__ cols,
    float* __restrict__ agg) {
  const int wv = __builtin_amdgcn_readfirstlane((int)(threadIdx.x >> 5));
  const int e  = blockIdx.x * 8 + wv;            // one edge per wave
  if (e >= N_EDGES) return;
  const unsigned p = (threadIdx.x & 31u) * 2u;   // channel pair per lane
  const int   c = cols[e];                       // wave-uniform -> scalar loads
  const int   r = rows[e];
  const float v = vals[e];
  const float2 hv = *(const float2*)(H + (size_t)c * N_CH + p);
  float* op = agg + (size_t)r * N_CH + p;
  __hip_atomic_fetch_add(op,     v * hv.x, __ATOMIC_RELAXED, __HIP_MEMORY_SCOPE_AGENT);
  __hip_atomic_fetch_add(op + 1, v * hv.y, __ATOMIC_RELAXED, __HIP_MEMORY_SCOPE_AGENT);
}

// ---------------------------------------------------------------------------
// Kernel 4: out = selu(agg + skip_weight)
// ---------------------------------------------------------------------------
__global__ __launch_bounds__(256) void selu_epilogue(
    const float* __restrict__ agg, const float* __restrict__ skip,
    float* __restrict__ out) {
  unsigned t = blockIdx.x * 256u + threadIdx.x;
  if (t >= (unsigned)N_NODES * N_CH) return;
  const float x = agg[t] + skip[t & (N_CH - 1)];
  const float alpha = 1.6732632423543772f;
  const float scale = 1.0507009873554805f;
  const float neg = scale * alpha * (expf(x) - 1.0f);
  const float pos = scale * x;
  out[t] = (x > 0.0f) ? pos : neg;
}

// ---------------------------------------------------------------------------
extern "C" void kernel_launch(void* const* d_in, const int* in_sizes, int n_in,
                              void* d_out, int out_size, void* d_ws, size_t ws_size,
                              hipStream_t stream) {
  const float* features = (const float*)d_in[0];  // [100000, 64]
  const float* kernelW  = (const float*)d_in[1];  // [64, 64]
  const float* bias     = (const float*)d_in[2];  // [64]
  const float* skip     = (const float*)d_in[3];  // [64]
  const float* vals     = (const float*)d_in[4];  // [E]
  const int*   rows     = (const int*)d_in[5];    // [E]
  const int*   cols     = (const int*)d_in[6];    // [E]

  float* out = (float*)d_out;
  float* h   = out;             // stage h in d_out (same size), overwritten later
  float* agg = (float*)d_ws;    // 100000*64 floats = 25.6 MB scratch

  // zero agg (must be re-done every call; harness does not re-poison)
  const int n4 = N_NODES * N_CH / 4;
  zero_f4<<<(n4 + 255) / 256, 256, 0, stream>>>((float4*)agg, n4);

  // h = F @ W + bias
  const int gemm_blocks = (N_NODES + 127) / 128;  // 8 waves x 16 rows per block
  gcn_gemm_wmma<<<gemm_blocks, 256, 0, stream>>>(features, kernelW, bias, h);

  // agg[row] += val * h[col]   (one wave per edge)
  const int sc_blocks = (N_EDGES + 7) / 8;        // 8 waves (edges) per block
  spmm_scatter<<<sc_blocks, 256, 0, stream>>>(h, vals, rows, cols, agg);

  // out = selu(agg + skip)
  const int ep_blocks = (N_NODES * N_CH + 255) / 256;
  selu_epilogue<<<ep_blocks, 256, 0, stream>>>(agg, skip, out);
}